// ADMMStiglerRouteB_47433618817423
// MI455X (gfx1250) — compile-verified
//
#include <hip/hip_runtime.h>

typedef __attribute__((ext_vector_type(2))) float v2f;
typedef __attribute__((ext_vector_type(8))) float v8f;

#define M_DIM 2048
#define N_DIM 4096
#define NBLOCKS 128
#define TPB 256
#define ADMM_ITERS 100
#define PGD_INNER 20
#define RHO_C 1.0f
#define STEP_C 5e-5f

// ---------------- grid-wide sense barrier (persistent kernel) ----------------
__device__ __forceinline__ void grid_barrier(unsigned* cnt, unsigned* gen) {
  __syncthreads();
  __threadfence();
  if (threadIdx.x == 0) {
    unsigned g = __hip_atomic_load(gen, __ATOMIC_ACQUIRE, __HIP_MEMORY_SCOPE_AGENT);
    if (__hip_atomic_fetch_add(cnt, 1u, __ATOMIC_ACQ_REL, __HIP_MEMORY_SCOPE_AGENT) ==
        (unsigned)(NBLOCKS - 1)) {
      __hip_atomic_store(cnt, 0u, __ATOMIC_RELAXED, __HIP_MEMORY_SCOPE_AGENT);
      __hip_atomic_fetch_add(gen, 1u, __ATOMIC_ACQ_REL, __HIP_MEMORY_SCOPE_AGENT);
    } else {
      while (__hip_atomic_load(gen, __ATOMIC_ACQUIRE, __HIP_MEMORY_SCOPE_AGENT) == g) {
        __builtin_amdgcn_s_sleep(2);
      }
    }
  }
  __syncthreads();
  __threadfence();
}

// ---------------- zacc += A * x  (16-row block per wave, K-segmented) --------
// WMMA f32 16x16x4: A-operand lane m: {A[m][k], A[m][k+1]}, lane m+16: k+2,k+3.
// B-operand broadcasts x across all 16 columns -> every D column == partial y.
__device__ __forceinline__ void pass_Ax(const float* __restrict__ A,
                                        const float* __restrict__ xv,
                                        float* __restrict__ zacc,
                                        int wave, int lane) {
  const int rowblk = wave & 127;            // 128 blocks x 16 rows = 2048 rows
  const int kseg   = wave >> 7;             // 8 K-segments of 512
  const int r0 = rowblk << 4;
  const int k0 = kseg * (N_DIM / 8);
  const int k1 = k0 + (N_DIM / 8);
  const int half = lane >> 4;
  const int mrow = lane & 15;
  const float* Arow = A + (size_t)(r0 + mrow) * N_DIM;

  v8f acc0 = {};
  v8f acc1 = {};
  for (int k = k0; k < k1; k += 8) {
    __builtin_prefetch(&Arow[k + 256], 0, 1);
    v2f a0, b0, a1, b1;
    a0.x = Arow[k + 2 * half];
    a0.y = Arow[k + 2 * half + 1];
    b0.x = xv[k + 2 * half];
    b0.y = xv[k + 2 * half + 1];
    a1.x = Arow[k + 4 + 2 * half];
    a1.y = Arow[k + 4 + 2 * half + 1];
    b1.x = xv[k + 4 + 2 * half];
    b1.y = xv[k + 4 + 2 * half + 1];
    acc0 = __builtin_amdgcn_wmma_f32_16x16x4_f32(false, a0, false, b0, (short)0,
                                                 acc0, false, false);
    acc1 = __builtin_amdgcn_wmma_f32_16x16x4_f32(false, a1, false, b1, (short)0,
                                                 acc1, false, false);
  }
  v8f acc = acc0 + acc1;
  // D layout: VGPR g holds M=g (lanes 0-15) and M=8+g (lanes 16-31); all
  // columns identical -> lanes 0 and 16 carry the 16 row partials.
  if (mrow == 0) {
    const int base = r0 + 8 * half;
#pragma unroll
    for (int g = 0; g < 8; ++g) atomicAdd(&zacc[base + g], acc[g]);
  }
}

// ---------------- wacc += A^T * t  (16-col block per wave, I-segmented) ------
__device__ __forceinline__ void pass_ATt(const float* __restrict__ A,
                                         const float* __restrict__ tv,
                                         float* __restrict__ wacc,
                                         int wave, int lane) {
  const int colblk = wave & 255;            // 256 blocks x 16 cols = 4096 cols
  const int iseg   = wave >> 8;             // 4 I-segments of 512
  const int j0 = colblk << 4;
  const int i0 = iseg * (M_DIM / 4);
  const int i1 = i0 + (M_DIM / 4);
  const int half = lane >> 4;
  const int mcol = lane & 15;               // M dimension = columns of A

  v8f acc0 = {};
  v8f acc1 = {};
  for (int i = i0; i < i1; i += 8) {
    __builtin_prefetch(&A[(size_t)(i + 64) * N_DIM + j0 + mcol], 0, 1);
    v2f a0, b0, a1, b1;
    a0.x = A[(size_t)(i + 2 * half) * N_DIM + j0 + mcol];
    a0.y = A[(size_t)(i + 2 * half + 1) * N_DIM + j0 + mcol];
    b0.x = tv[i + 2 * half];
    b0.y = tv[i + 2 * half + 1];
    a1.x = A[(size_t)(i + 4 + 2 * half) * N_DIM + j0 + mcol];
    a1.y = A[(size_t)(i + 4 + 2 * half + 1) * N_DIM + j0 + mcol];
    b1.x = tv[i + 4 + 2 * half];
    b1.y = tv[i + 4 + 2 * half + 1];
    acc0 = __builtin_amdgcn_wmma_f32_16x16x4_f32(false, a0, false, b0, (short)0,
                                                 acc0, false, false);
    acc1 = __builtin_amdgcn_wmma_f32_16x16x4_f32(false, a1, false, b1, (short)0,
                                                 acc1, false, false);
  }
  v8f acc = acc0 + acc1;
  if (mcol == 0) {
    const int base = j0 + 8 * half;
#pragma unroll
    for (int g = 0; g < 8; ++g) atomicAdd(&wacc[base + g], acc[g]);
  }
}

// ---------------- persistent ADMM kernel ------------------------------------
__global__ __launch_bounds__(TPB) void admm_persistent(
    const float* __restrict__ A, const float* __restrict__ bvec,
    const float* __restrict__ cvec, float* __restrict__ out,
    unsigned char* __restrict__ wsb) {
  unsigned* cnt = (unsigned*)wsb;
  unsigned* gen = cnt + 1;
  float* fws  = (float*)(wsb + 256);
  float* xv   = fws;               // N
  float* zbuf = xv + N_DIM;        // M  (preloaded with -r so pass_Ax yields Ax-r)
  float* w0   = zbuf + M_DIM;      // N  (ping-pong A^T accumulators)
  float* w1   = w0 + N_DIM;        // N
  float* rv   = w1 + N_DIM;        // M
  float* svec = rv + M_DIM;        // M
  float* uvec = svec + M_DIM;      // M

  const int t = blockIdx.x * TPB + (int)threadIdx.x;
  const int wave = t >> 5;
  const int lane = (int)threadIdx.x & 31;

  // init: x=s=u=0, r=b, zbuf=-r, w buffers zero
  if (t < N_DIM) { xv[t] = 0.f; w0[t] = 0.f; w1[t] = 0.f; }
  if (t < M_DIM) {
    float bb = bvec[t];
    svec[t] = 0.f; uvec[t] = 0.f; rv[t] = bb; zbuf[t] = -bb;
  }
  grid_barrier(cnt, gen);

  for (int outer = 0; outer < ADMM_ITERS; ++outer) {
    for (int inner = 0; inner < PGD_INNER; ++inner) {
      float* wcur  = (inner & 1) ? w1 : w0;
      float* wnext = (inner & 1) ? w0 : w1;

      // zbuf (== -r) += A x   ->  zbuf = A x - r
      pass_Ax(A, xv, zbuf, wave, lane);
      grid_barrier(cnt, gen);

      // wcur += A^T (A x - r)
      pass_ATt(A, zbuf, wcur, wave, lane);
      grid_barrier(cnt, gen);

      // x = max(x - STEP*(c + RHO*w), 0); re-arm buffers for next step
      if (t < N_DIM) {
        float g  = cvec[t] + RHO_C * wcur[t];
        float xn = xv[t] - STEP_C * g;
        xv[t] = xn > 0.f ? xn : 0.f;
        wnext[t] = 0.f;
      }
      if (t < M_DIM) zbuf[t] = (inner == PGD_INNER - 1) ? 0.f : -rv[t];
      grid_barrier(cnt, gen);
    }

    // final z = A x for this outer iteration (zbuf preset to 0 above)
    pass_Ax(A, xv, zbuf, wave, lane);
    grid_barrier(cnt, gen);

    if (t < M_DIM) {
      float zi = zbuf[t], bb = bvec[t], ui = uvec[t];
      float s_ = zi - bb + ui;
      s_ = s_ > 0.f ? s_ : 0.f;
      float un = ui + zi - s_ - bb;
      float rn = s_ + bb - un;
      svec[t] = s_; uvec[t] = un; rv[t] = rn;
      zbuf[t] = -rn;               // pre-arm for next outer's first PGD step
    }
    grid_barrier(cnt, gen);
  }

  // outputs: concat(x[N], s[M], u[M], lambda=-RHO*u [M])
  if (t < N_DIM) out[t] = xv[t];
  if (t < M_DIM) {
    out[N_DIM + t]             = svec[t];
    out[N_DIM + M_DIM + t]     = uvec[t];
    out[N_DIM + 2 * M_DIM + t] = -RHO_C * uvec[t];
  }
}

extern "C" void kernel_launch(void* const* d_in, const int* in_sizes, int n_in,
                              void* d_out, int out_size, void* d_ws,
                              size_t ws_size, hipStream_t stream) {
  (void)in_sizes; (void)n_in; (void)out_size; (void)ws_size;
  const float* A = (const float*)d_in[0];
  const float* b = (const float*)d_in[1];
  const float* c = (const float*)d_in[2];
  float* out = (float*)d_out;
  // reset the grid-barrier counters each call (graph-capture safe, deterministic)
  hipMemsetAsync(d_ws, 0, 256, stream);
  admm_persistent<<<NBLOCKS, TPB, 0, stream>>>(A, b, c, out,
                                               (unsigned char*)d_ws);
}